// TracableDecodeV10_66709432041890
// MI455X (gfx1250) — compile-verified
//
#include <hip/hip_runtime.h>
#include <hip/hip_bf16.h>
#include <math.h>

#define L_LAYERS 28
#define H 1024
#define NH 8
#define KVH 4
#define HD 128
#define I_DIM 3072
#define VC 4096
#define NCG 16
#define B 4
#define S 512
#define SP1 513
#define THETA 1000000.0f
#define EPS 1e-6f

typedef __attribute__((ext_vector_type(2))) float v2f;
typedef __attribute__((ext_vector_type(8))) float v8f;

// ---------------------------------------------------------------------------
// KV-cache re-layout: (2L,B,KVH,S,HD) -> (2L,B,KVH,S+1,HD), slot S left for new k/v
// ---------------------------------------------------------------------------
__global__ __launch_bounds__(256) void cache_copy_kernel(const float4* __restrict__ src,
                                                         float4* __restrict__ dst) {
  const size_t chunk4  = (size_t)S * HD / 4;    // 16384 float4 per (l2,b,g)
  const size_t dchunk4 = (size_t)SP1 * HD / 4;  // 16416
  const size_t nchunk  = (size_t)2 * L_LAYERS * B * KVH;
  const size_t total   = chunk4 * nchunk;
  for (size_t i = (size_t)blockIdx.x * blockDim.x + threadIdx.x; i < total;
       i += (size_t)gridDim.x * blockDim.x) {
    size_t c = i / chunk4;
    size_t r = i - c * chunk4;
    dst[c * dchunk4 + r] = src[i];
  }
}

// ---------------------------------------------------------------------------
// Embedding: h[b][j] = cb0_emb[id0[b]][j] + sum_g cp_emb[g][idg][j] + trailing[b][j]
// ---------------------------------------------------------------------------
__global__ __launch_bounds__(256) void embed_kernel(const int* __restrict__ cb0_id,
                                                    const int* __restrict__ cb1_ids,
                                                    const float* __restrict__ trailing,
                                                    const float* __restrict__ cb0_emb,
                                                    const float* __restrict__ cp_emb,
                                                    float* __restrict__ h) {
  int b = blockIdx.x;
  int id0 = cb0_id[b];
  for (int j = threadIdx.x; j < H; j += blockDim.x) {
    float acc = cb0_emb[(size_t)id0 * H + j] + trailing[(size_t)b * H + j];
    for (int g = 0; g < NCG - 1; ++g) {
      int idg = cb1_ids[b * (NCG - 1) + g];
      acc += cp_emb[((size_t)g * VC + idg) * H + j];
    }
    h[(size_t)b * H + j] = acc;
  }
}

// ---------------------------------------------------------------------------
// RMSNorm over last dim n (one block per batch row)
// ---------------------------------------------------------------------------
__global__ __launch_bounds__(256) void rmsnorm_kernel(const float* __restrict__ in,
                                                      const float* __restrict__ w,
                                                      float* __restrict__ outv, int n) {
  __shared__ float red[256];
  int b = blockIdx.x;
  const float* x = in + (size_t)b * n;
  float s = 0.f;
  for (int j = threadIdx.x; j < n; j += 256) { float v = x[j]; s += v * v; }
  red[threadIdx.x] = s;
  __syncthreads();
  for (int st = 128; st > 0; st >>= 1) {
    if (threadIdx.x < st) red[threadIdx.x] += red[threadIdx.x + st];
    __syncthreads();
  }
  float scl = rsqrtf(red[0] / (float)n + EPS);
  for (int j = threadIdx.x; j < n; j += 256)
    outv[(size_t)b * n + j] = x[j] * scl * w[j];
}

// ---------------------------------------------------------------------------
// GEMV via V_WMMA_F32_16X16X4_F32, K-unrolled x2 + weight-stream prefetch.
// out[m][n] (m<4) = (base? base : 0) + sum_k X[m][k]*W[k][n],  W row-major KxN.
// Block: 256 thr (8 waves). Each wave: same 16-col tile, 1/8 of K. LDS reduce.
// Dyn LDS: 4*K floats (X stage) + 512 floats (partials).
// ---------------------------------------------------------------------------
__global__ __launch_bounds__(256) void gemv_wmma_kernel(const float* __restrict__ X,
                                                        const float* __restrict__ W,
                                                        const float* __restrict__ base,
                                                        float* __restrict__ out,
                                                        int K, int N) {
  extern __shared__ float smem[];
  float* Xs = smem;            // 4*K  (B rows contiguous)
  float* Ps = smem + 4 * K;    // 8 waves * 4 rows * 16 cols
  int tid = threadIdx.x;
  for (int i = tid; i < 4 * K; i += 256) Xs[i] = X[i];
  __syncthreads();

  int lane = tid & 31;
  int wave = tid >> 5;
  int col  = lane & 15;        // A: M index / B: N index
  int hlf  = lane >> 4;        // half-wave selector
  int n0   = blockIdx.x * 16;
  int kchunk = K >> 3;         // per-wave K span (multiple of 8 for all our K)
  int kb = wave * kchunk;

  const float* wp = W + n0 + col;
  v8f c = {};
  for (int k = kb; k < kb + kchunk; k += 8) {
    // prefetch the weight rows one unrolled iteration ahead (global_prefetch_b8)
    __builtin_prefetch(wp + (size_t)(k + 8 + hlf) * N, 0, 1);
    __builtin_prefetch(wp + (size_t)(k + 12 + hlf) * N, 0, 1);

    v2f a0, b0, a1, b1;
    // A frag (16x4 f32): lanes 0-15 -> K = k,k+1 ; lanes 16-31 -> K = k+2,k+3
    int ka = k + hlf * 2;
    if (col < 4) {
      a0.x = Xs[col * K + ka];     a0.y = Xs[col * K + ka + 1];
      a1.x = Xs[col * K + ka + 4]; a1.y = Xs[col * K + ka + 5];
    } else {
      a0.x = 0.f; a0.y = 0.f; a1.x = 0.f; a1.y = 0.f;
    }
    // B frags (4x16 f32): four 64B row segments — this IS the weight stream
    b0.x = wp[(size_t)(k + hlf) * N];
    b0.y = wp[(size_t)(k + 2 + hlf) * N];
    b1.x = wp[(size_t)(k + 4 + hlf) * N];
    b1.y = wp[(size_t)(k + 6 + hlf) * N];

    c = __builtin_amdgcn_wmma_f32_16x16x4_f32(false, a0, false, b0,
                                              (short)0, c, false, false);
    c = __builtin_amdgcn_wmma_f32_16x16x4_f32(false, a1, false, b1,
                                              (short)0, c, false, false);
  }
  // C layout: lanes 0-15 hold rows M=0..7 in c[0..7]; we need rows 0..3
  if (lane < 16) {
    Ps[(wave * 4 + 0) * 16 + col] = c[0];
    Ps[(wave * 4 + 1) * 16 + col] = c[1];
    Ps[(wave * 4 + 2) * 16 + col] = c[2];
    Ps[(wave * 4 + 3) * 16 + col] = c[3];
  }
  __syncthreads();
  if (tid < 64) {
    int m  = tid >> 4;
    int cc = tid & 15;
    float acc = 0.f;
    for (int w2 = 0; w2 < 8; ++w2) acc += Ps[(w2 * 4 + m) * 16 + cc];
    size_t oi = (size_t)m * N + n0 + cc;
    if (base) acc += base[oi];
    out[oi] = acc;
  }
}

// ---------------------------------------------------------------------------
// Per-head q/k rmsnorm + RoPE, and write new k/v into output cache at s=S.
// grid = B*16, block = 128.  hh<8: q head, 8..11: k head, 12..15: v copy.
// ---------------------------------------------------------------------------
__global__ __launch_bounds__(128) void rope_kernel(float* __restrict__ q,
                                                   float* __restrict__ k,
                                                   const float* __restrict__ v,
                                                   const float* __restrict__ qnw,
                                                   const float* __restrict__ knw,
                                                   const int* __restrict__ position,
                                                   float* __restrict__ kc_dst,
                                                   float* __restrict__ vc_dst) {
  __shared__ float red[128];
  __shared__ float xn[128];
  int b  = blockIdx.x >> 4;
  int hh = blockIdx.x & 15;
  int d  = threadIdx.x;
  float pos = (float)position[b];
  float ang = pos * __powf(THETA, -(float)(2 * (d & 63)) / (float)HD);
  float cs = __cosf(ang), sn = __sinf(ang);

  if (hh < 8) {
    float* vec = q + ((size_t)b * NH + hh) * HD;
    float x = vec[d];
    red[d] = x * x; __syncthreads();
    for (int st = 64; st > 0; st >>= 1) { if (d < st) red[d] += red[d + st]; __syncthreads(); }
    float xv = x * rsqrtf(red[0] / (float)HD + EPS) * qnw[d];
    xn[d] = xv; __syncthreads();
    float rot = (d < 64) ? -xn[d + 64] : xn[d - 64];
    vec[d] = xv * cs + rot * sn;
  } else if (hh < 12) {
    int g = hh - 8;
    const float* vec = k + ((size_t)b * KVH + g) * HD;
    float x = vec[d];
    red[d] = x * x; __syncthreads();
    for (int st = 64; st > 0; st >>= 1) { if (d < st) red[d] += red[d + st]; __syncthreads(); }
    float xv = x * rsqrtf(red[0] / (float)HD + EPS) * knw[d];
    xn[d] = xv; __syncthreads();
    float rot = (d < 64) ? -xn[d + 64] : xn[d - 64];
    kc_dst[(((size_t)b * KVH + g) * SP1 + S) * HD + d] = xv * cs + rot * sn;
  } else {
    int g = hh - 12;
    vc_dst[(((size_t)b * KVH + g) * SP1 + S) * HD + d] =
        v[((size_t)b * KVH + g) * HD + d];
  }
}

// ---------------------------------------------------------------------------
// GQA attention (decode): one block per (b, kv-group); 2 q-heads per group.
// ---------------------------------------------------------------------------
__global__ __launch_bounds__(256) void attn_kernel(const float* __restrict__ q,
                                                   const float* __restrict__ kc,
                                                   const float* __restrict__ vc,
                                                   float* __restrict__ o) {
  __shared__ float qs[2][HD];
  __shared__ float sc[2][SP1];
  __shared__ float red[256];
  const float scale = 0.088388347648318447f;  // 1/sqrt(128)
  int b = blockIdx.x / KVH;
  int g = blockIdx.x % KVH;
  int tid = threadIdx.x;
  {
    int r = tid >> 7, d = tid & 127;
    qs[r][d] = q[((size_t)b * NH + g * 2 + r) * HD + d];
  }
  __syncthreads();

  int wave = tid >> 5, lane = tid & 31;
  const float* kg = kc + ((size_t)b * KVH + g) * SP1 * HD;
  for (int s = wave; s < SP1; s += 8) {
    float4 kv4 = ((const float4*)(kg + (size_t)s * HD))[lane];
    int d4 = lane * 4;
    float d0 = kv4.x * qs[0][d4] + kv4.y * qs[0][d4 + 1] +
               kv4.z * qs[0][d4 + 2] + kv4.w * qs[0][d4 + 3];
    float d1 = kv4.x * qs[1][d4] + kv4.y * qs[1][d4 + 1] +
               kv4.z * qs[1][d4 + 2] + kv4.w * qs[1][d4 + 3];
    for (int off = 16; off > 0; off >>= 1) {
      d0 += __shfl_down(d0, off);
      d1 += __shfl_down(d1, off);
    }
    if (lane == 0) { sc[0][s] = d0 * scale; sc[1][s] = d1 * scale; }
  }
  __syncthreads();

  for (int r = 0; r < 2; ++r) {
    float m = -1e30f;
    for (int s = tid; s < SP1; s += 256) m = fmaxf(m, sc[r][s]);
    red[tid] = m; __syncthreads();
    for (int st = 128; st > 0; st >>= 1) {
      if (tid < st) red[tid] = fmaxf(red[tid], red[tid + st]);
      __syncthreads();
    }
    m = red[0]; __syncthreads();
    float sum = 0.f;
    for (int s = tid; s < SP1; s += 256) {
      float e = __expf(sc[r][s] - m);
      sc[r][s] = e;
      sum += e;
    }
    red[tid] = sum; __syncthreads();
    for (int st = 128; st > 0; st >>= 1) {
      if (tid < st) red[tid] += red[tid + st];
      __syncthreads();
    }
    float inv = 1.f / red[0]; __syncthreads();
    for (int s = tid; s < SP1; s += 256) sc[r][s] *= inv;
    __syncthreads();
  }

  const float* vg = vc + ((size_t)b * KVH + g) * SP1 * HD;
  int r = tid >> 7, d = tid & 127;
  float acc = 0.f;
  for (int s = 0; s < SP1; ++s) acc += sc[r][s] * vg[(size_t)s * HD + d];
  o[((size_t)b * NH + g * 2 + r) * HD + d] = acc;
}

// ---------------------------------------------------------------------------
__global__ __launch_bounds__(256) void silu_mul_kernel(const float* __restrict__ g,
                                                       const float* __restrict__ u,
                                                       float* __restrict__ m, int n) {
  int i = blockIdx.x * 256 + threadIdx.x;
  if (i < n) {
    float gv = g[i];
    m[i] = (gv / (1.f + __expf(-gv))) * u[i];
  }
}

// ---------------------------------------------------------------------------
extern "C" void kernel_launch(void* const* d_in, const int* in_sizes, int n_in,
                              void* d_out, int out_size, void* d_ws, size_t ws_size,
                              hipStream_t stream) {
  const int*   cb0_id   = (const int*)d_in[0];
  const int*   cb1_ids  = (const int*)d_in[1];
  const float* trailing = (const float*)d_in[2];
  const float* kv_in    = (const float*)d_in[3];
  const int*   position = (const int*)d_in[4];
  const float* cb0_emb  = (const float*)d_in[5];
  const float* cp_emb   = (const float*)d_in[6];
  const float* Wq_all   = (const float*)d_in[7];
  const float* Wk_all   = (const float*)d_in[8];
  const float* Wv_all   = (const float*)d_in[9];
  const float* Wo_all   = (const float*)d_in[10];
  const float* qnw_all  = (const float*)d_in[11];
  const float* knw_all  = (const float*)d_in[12];
  const float* ln1_all  = (const float*)d_in[13];
  const float* ln2_all  = (const float*)d_in[14];
  const float* Wg_all   = (const float*)d_in[15];
  const float* Wu_all   = (const float*)d_in[16];
  const float* Wd_all   = (const float*)d_in[17];
  const float* norm_w   = (const float*)d_in[18];
  const float* head_w   = (const float*)d_in[19];

  float* out    = (float*)d_out;
  float* logits = out;                                   // B*VC
  float* cache  = out + (size_t)B * VC;                  // 2L*B*KVH*SP1*HD
  float* ph     = cache + (size_t)2 * L_LAYERS * B * KVH * SP1 * HD;  // B*H

  float* h = (float*)d_ws;
  float* x = h + (size_t)B * H;
  float* q = x + (size_t)B * H;
  float* k = q + (size_t)B * NH * HD;
  float* v = k + (size_t)B * KVH * HD;
  float* o = v + (size_t)B * KVH * HD;
  float* g = o + (size_t)B * NH * HD;
  float* u = g + (size_t)B * I_DIM;

  const size_t smemH = (4 * (size_t)H + 512) * sizeof(float);
  const size_t smemI = (4 * (size_t)I_DIM + 512) * sizeof(float);

  cache_copy_kernel<<<4096, 256, 0, stream>>>((const float4*)kv_in, (float4*)cache);
  embed_kernel<<<B, 256, 0, stream>>>(cb0_id, cb1_ids, trailing, cb0_emb, cp_emb, h);

  for (int l = 0; l < L_LAYERS; ++l) {
    const float* Wq = Wq_all + (size_t)l * H * (NH * HD);
    const float* Wk = Wk_all + (size_t)l * H * (KVH * HD);
    const float* Wv = Wv_all + (size_t)l * H * (KVH * HD);
    const float* Wo = Wo_all + (size_t)l * (NH * HD) * H;
    const float* Wg = Wg_all + (size_t)l * H * I_DIM;
    const float* Wu = Wu_all + (size_t)l * H * I_DIM;
    const float* Wd = Wd_all + (size_t)l * I_DIM * H;
    float* kc = cache + (size_t)(2 * l) * B * KVH * SP1 * HD;
    float* vc = kc + (size_t)B * KVH * SP1 * HD;

    rmsnorm_kernel<<<B, 256, 0, stream>>>(h, ln1_all + (size_t)l * H, x, H);
    gemv_wmma_kernel<<<(NH * HD) / 16, 256, smemH, stream>>>(x, Wq, nullptr, q, H, NH * HD);
    gemv_wmma_kernel<<<(KVH * HD) / 16, 256, smemH, stream>>>(x, Wk, nullptr, k, H, KVH * HD);
    gemv_wmma_kernel<<<(KVH * HD) / 16, 256, smemH, stream>>>(x, Wv, nullptr, v, H, KVH * HD);
    rope_kernel<<<B * 16, 128, 0, stream>>>(q, k, v, qnw_all + (size_t)l * HD,
                                            knw_all + (size_t)l * HD, position, kc, vc);
    attn_kernel<<<B * KVH, 256, 0, stream>>>(q, kc, vc, o);
    gemv_wmma_kernel<<<H / 16, 256, smemH, stream>>>(o, Wo, h, h, NH * HD, H);
    rmsnorm_kernel<<<B, 256, 0, stream>>>(h, ln2_all + (size_t)l * H, x, H);
    gemv_wmma_kernel<<<I_DIM / 16, 256, smemH, stream>>>(x, Wg, nullptr, g, H, I_DIM);
    gemv_wmma_kernel<<<I_DIM / 16, 256, smemH, stream>>>(x, Wu, nullptr, u, H, I_DIM);
    silu_mul_kernel<<<(B * I_DIM + 255) / 256, 256, 0, stream>>>(g, u, g, B * I_DIM);
    gemv_wmma_kernel<<<H / 16, 256, smemI, stream>>>(g, Wd, h, h, I_DIM, H);
  }

  rmsnorm_kernel<<<B, 256, 0, stream>>>(h, norm_w, ph, H);
  gemv_wmma_kernel<<<VC / 16, 256, smemH, stream>>>(ph, head_w, nullptr, logits, H, VC);
}